// SparseG_iiComputation_62319975465265
// MI455X (gfx1250) — compile-verified
//
#include <hip/hip_runtime.h>
#include <hip/hip_bf16.h>

typedef __attribute__((ext_vector_type(2))) float v2f;
typedef __attribute__((ext_vector_type(8))) float v8f;

#define B_DIM 64
#define N_DIM 8192
#define SEG   32      // elements per thread in scan
#define SCAN_T 256    // threads per scan block

// -------- async global->LDS path (guarded: falls back to plain loads) ------
#if defined(__gfx1250__) && __has_builtin(__builtin_amdgcn_global_load_async_to_lds_b128)
#define USE_ASYNC_LDS 1
typedef __attribute__((ext_vector_type(4))) int i32x4;
typedef __attribute__((address_space(1))) i32x4 gi32x4;   // global int4
typedef __attribute__((address_space(3))) i32x4 li32x4;   // LDS int4
__device__ __forceinline__ void wait_asynccnt0() {
#if __has_builtin(__builtin_amdgcn_s_wait_asynccnt)
    __builtin_amdgcn_s_wait_asynccnt(0);
#else
    asm volatile("s_wait_asynccnt 0" ::: "memory");
#endif
}
#else
#define USE_ASYNC_LDS 0
#endif

// ---------------- complex helpers ----------------
struct cplx { float x, y; };
__device__ __forceinline__ cplx cmul(cplx a, cplx b) {
    return {a.x * b.x - a.y * b.y, a.x * b.y + a.y * b.x};
}
__device__ __forceinline__ cplx cadd(cplx a, cplx b) { return {a.x + b.x, a.y + b.y}; }
__device__ __forceinline__ cplx csub(cplx a, cplx b) { return {a.x - b.x, a.y - b.y}; }
__device__ __forceinline__ cplx crecip(cplx a) {
    float d = 1.0f / (a.x * a.x + a.y * a.y);
    return {a.x * d, -a.y * d};
}
__device__ __forceinline__ cplx cdiv(cplx n, cplx d) {
    float s = 1.0f / (d.x * d.x + d.y * d.y);
    return {(n.x * d.x + n.y * d.y) * s, (n.y * d.x - n.x * d.y) * s};
}

// 2x2 complex Moebius matrix [[a,b],[c,d]]
struct cmat { cplx a, b, c, d; };
__device__ __forceinline__ cmat mident() {
    return {{1.f, 0.f}, {0.f, 0.f}, {0.f, 0.f}, {1.f, 0.f}};
}
__device__ __forceinline__ cmat mmul(const cmat& X, const cmat& Y) {
    cmat r;
    r.a = cadd(cmul(X.a, Y.a), cmul(X.b, Y.c));
    r.b = cadd(cmul(X.a, Y.b), cmul(X.b, Y.d));
    r.c = cadd(cmul(X.c, Y.a), cmul(X.d, Y.c));
    r.d = cadd(cmul(X.c, Y.b), cmul(X.d, Y.d));
    return r;
}
// Moebius transforms are scale-invariant: renormalize to avoid fp32 overflow
__device__ __forceinline__ void mnorm(cmat& M) {
    float mx = fabsf(M.a.x);
    mx = fmaxf(mx, fabsf(M.a.y)); mx = fmaxf(mx, fabsf(M.b.x)); mx = fmaxf(mx, fabsf(M.b.y));
    mx = fmaxf(mx, fabsf(M.c.x)); mx = fmaxf(mx, fabsf(M.c.y));
    mx = fmaxf(mx, fabsf(M.d.x)); mx = fmaxf(mx, fabsf(M.d.y));
    float s = 1.0f / fmaxf(mx, 1e-30f);
    M.a.x *= s; M.a.y *= s; M.b.x *= s; M.b.y *= s;
    M.c.x *= s; M.c.y *= s; M.d.x *= s; M.d.y *= s;
}
// left-multiply by elementary E(w) = [[w,-1],[1,0]]:  M <- E*M
__device__ __forceinline__ void estep(cmat& M, cplx w) {
    cplx na = csub(cmul(w, M.a), M.c);
    cplx nb = csub(cmul(w, M.b), M.d);
    M.c = M.a; M.d = M.b; M.a = na; M.b = nb;
}

__device__ __forceinline__ void mstore(float* sm, int idx, const cmat& M) {
    float* p = sm + idx * 9;  // pad stride 9 -> no LDS bank conflicts
    p[0] = M.a.x; p[1] = M.a.y; p[2] = M.b.x; p[3] = M.b.y;
    p[4] = M.c.x; p[5] = M.c.y; p[6] = M.d.x; p[7] = M.d.y;
}
__device__ __forceinline__ cmat mload(const float* sm, int idx) {
    const float* p = sm + idx * 9;
    cmat M;
    M.a = {p[0], p[1]}; M.b = {p[2], p[3]};
    M.c = {p[4], p[5]}; M.d = {p[6], p[7]};
    return M;
}

// ============================================================================
// Kernel 1: bidirectional continued-fraction scan (Moebius blocked scan)
// One block per batch row; 256 threads x 32 elements.
// ============================================================================
__global__ __launch_bounds__(SCAN_T) void gii_scan_kernel(
    const float* __restrict__ v, const float* __restrict__ mask,
    float* __restrict__ out_sparse,  // (B,N,2)
    float* __restrict__ xws)         // (B,2,N) feats_sparse for conv stage
{
    __shared__ float buf[N_DIM];          // 32 KB staging (v, then mask)
    __shared__ float sm[SCAN_T * 9];      // 9 KB scan matrices (padded)

    const int row  = blockIdx.x;
    const int tid  = threadIdx.x;
    const int base = tid * SEG;

    // stage v coalesced (async direct-to-LDS when available)
#if USE_ASYNC_LDS
    for (int c = tid; c < N_DIM / 4; c += SCAN_T)
        __builtin_amdgcn_global_load_async_to_lds_b128(
            (gi32x4*)(v + row * N_DIM + c * 4),
            (li32x4*)(buf + c * 4), 0, 0);
    wait_asynccnt0();
#else
    for (int i = tid; i < N_DIM; i += SCAN_T) buf[i] = v[row * N_DIM + i];
#endif
    __syncthreads();
    float a[SEG];
#pragma unroll
    for (int j = 0; j < SEG; ++j) a[j] = buf[base + j] - 2.0f;  // he_diag
    __syncthreads();
    // stage mask (consumed only after the barriers of both scans below)
#if USE_ASYNC_LDS
    for (int c = tid; c < N_DIM / 4; c += SCAN_T)
        __builtin_amdgcn_global_load_async_to_lds_b128(
            (gi32x4*)(mask + row * N_DIM + c * 4),
            (li32x4*)(buf + c * 4), 0, 0);
#else
    for (int i = tid; i < N_DIM; i += SCAN_T) buf[i] = mask[row * N_DIM + i];
#endif

    // ---------------- forward (L) ----------------
    cmat M = mident();
#pragma unroll
    for (int j = 0; j < SEG; ++j) { estep(M, {-a[j], 1.0f}); }
    mnorm(M);

    mstore(sm, tid, M);
    __syncthreads();
    for (int off = 1; off < SCAN_T; off <<= 1) {
        cmat other = mident();
        bool has = tid >= off;
        if (has) other = mload(sm, tid - off);
        __syncthreads();
        if (has) { M = mmul(M, other); mnorm(M); }
        mstore(sm, tid, M);
        __syncthreads();
    }
    cmat E = (tid > 0) ? mload(sm, tid - 1) : mident();

    // in-segment walk (stable direct recurrence L = w - 1/L_prev)
    cplx il = cdiv(E.c, E.a);  // 1/L_incoming (0 for tid==0)
    float Lre[SEG], Lim[SEG];
#pragma unroll
    for (int j = 0; j < SEG; ++j) {
        cplx w = {-a[j], 1.0f};
        cplx L = csub(w, il);
        Lre[j] = L.x; Lim[j] = L.y;
        il = crecip(L);
    }
    __syncthreads();  // done reading sm; safe to reuse

    // ---------------- backward (R) ----------------
    cmat S = mident();
#pragma unroll
    for (int j = SEG - 1; j >= 0; --j) { estep(S, {-a[j], 1.0f}); }
    mnorm(S);

    const int u = (SCAN_T - 1) - tid;  // reversed scan index
    mstore(sm, u, S);
    __syncthreads();
    for (int off = 1; off < SCAN_T; off <<= 1) {
        cmat other = mident();
        bool has = u >= off;
        if (has) other = mload(sm, u - off);
        __syncthreads();
        if (has) { S = mmul(S, other); mnorm(S); }
        mstore(sm, u, S);
        __syncthreads();
    }
    cmat Q = (u > 0) ? mload(sm, u - 1) : mident();

#if USE_ASYNC_LDS
    wait_asynccnt0();    // mask staging complete (per-wave), then barrier
#endif
    __syncthreads();

    // backward walk + combine: G = 1/(L + R - w), multiply by mask, write out
    cplx ir = cdiv(Q.c, Q.a);
#pragma unroll
    for (int j = SEG - 1; j >= 0; --j) {
        cplx w = {-a[j], 1.0f};
        cplx R = csub(w, ir);
        cplx D = {Lre[j] + R.x - w.x, Lim[j] + R.y - w.y};
        cplx G = crecip(D);
        float mk = buf[base + j];
        float fr = G.x * mk, fi = G.y * mk;
        int idx = base + j;
        int gi  = row * N_DIM + idx;
        out_sparse[2 * gi + 0] = fr;
        out_sparse[2 * gi + 1] = fi;
        xws[(row * 2 + 0) * N_DIM + idx] = fr;
        xws[(row * 2 + 1) * N_DIM + idx] = fi;
        ir = crecip(R);
    }
}

// ============================================================================
// Kernel 2: fused 3-layer Conv1d (2->32->32->2, K=5) + masked select.
// fp32 WMMA 16x16x4; one block = one batch row x 128-position tile; 4 waves.
// All fragment loads unconditional (zero-padded operand tiles in LDS).
// ============================================================================
#define TPOS 128              // output positions per block
#define YG   (TPOS + 16)      // y1/y2 grid: [n0-8, n0+T+8) -> 9 tiles of 16
#define XG   (TPOS + 32)      // x grid:    [n0-16, n0+T+16)
#define NTY  (YG / 16)        // 9
#define NTO  (TPOS / 16)      // 8

__device__ __forceinline__ v8f wmma_f32(v2f A, v2f Bv, v8f C) {
    return __builtin_amdgcn_wmma_f32_16x16x4_f32(
        false, A, false, Bv, (short)0, C, false, false);
}
// advance im2col (ci, kt) decomposition of flat K index by +4 without div/mod
__device__ __forceinline__ void kadv(int& ci, int& kt) {
    kt += 4;
    if (kt >= 5) { kt -= 5; ++ci; }
}

__global__ __launch_bounds__(128) void conv_interp_kernel(
    const float* __restrict__ mask,
    const float* __restrict__ w1, const float* __restrict__ b1,
    const float* __restrict__ w2, const float* __restrict__ b2,
    const float* __restrict__ w3, const float* __restrict__ b3,
    const float* __restrict__ xws,   // (B,2,N) feats_sparse
    float* __restrict__ out)          // (B,N,2) feats_final
{
    __shared__ float xs[3][XG];       // input tile; row 2 = zeros (K padding)
    __shared__ float y1s[32][YG];     // layer-1 activations
    __shared__ float y2s[32][YG];     // layer-2 activations
    __shared__ float w1s[32][12];     // zero-padded K 10->12
    __shared__ float w2s[32][160];
    __shared__ float w3s[2][160];
    __shared__ float b1s[32], b2s[32], b3s[2];

    const int tid  = threadIdx.x;
    const int wave = tid >> 5;
    const int lane = tid & 31;
    const bool hi  = lane >= 16;
    const int mrow = lane & 15;       // A row / B,C column index for this lane
    const int kof  = hi ? 2 : 0;      // hi half-wave holds K+2,K+3 slices

    const int row = blockIdx.y;
    const int n0  = blockIdx.x * TPOS;

    // cooperative weight + tile loads
    for (int i = tid; i < 320; i += 128)  w1s[i / 10][i % 10]   = w1[i];
    for (int i = tid; i < 5120; i += 128) w2s[i / 160][i % 160] = w2[i];
    for (int i = tid; i < 320; i += 128)  w3s[i / 160][i % 160] = w3[i];
    if (tid < 32) { b1s[tid] = b1[tid]; b2s[tid] = b2[tid];
                    w1s[tid][10] = 0.0f; w1s[tid][11] = 0.0f; }
    if (tid < 2)  { b3s[tid] = b3[tid]; }
    for (int i = tid; i < XG; i += 128) xs[2][i] = 0.0f;   // zero K-pad row
    for (int i = tid; i < 2 * XG; i += 128) {
        int c = i / XG, j = i % XG;
        int na = n0 - 16 + j;
        xs[c][j] = (na >= 0 && na < N_DIM) ? xws[(row * 2 + c) * N_DIM + na] : 0.0f;
    }
    __syncthreads();

    // ---- layer 1: 2->32, K-dim = 10 (zero-padded to 12 = 3 WMMA k-steps) ----
    for (int t = wave; t < NTY; t += 4) {
        const int gb = t * 16 + mrow;
        for (int cot = 0; cot < 2; ++cot) {
            const int co = cot * 16 + mrow;
            v8f acc = {};
            int k = kof, ci0 = 0, kt0 = kof, ci1 = 0, kt1 = kof + 1;
#pragma unroll
            for (int ks = 0; ks < 3; ++ks) {
                v2f A, Bv;
                A.x  = w1s[co][k];
                A.y  = w1s[co][k + 1];
                Bv.x = xs[ci0][gb + kt0 + 6];
                Bv.y = xs[ci1][gb + kt1 + 6];
                acc = wmma_f32(A, Bv, acc);
                k += 4; kadv(ci0, kt0); kadv(ci1, kt1);
            }
#pragma unroll
            for (int r = 0; r < 8; ++r) {
                int c2 = cot * 16 + r + (hi ? 8 : 0);
                y1s[c2][t * 16 + mrow] = fmaxf(acc[r] + b1s[c2], 0.0f);
            }
        }
    }
    __syncthreads();

    // ---- layer 2: 32->32, K-dim = 160 (40 WMMA k-steps) ----
    for (int t = wave; t < NTY; t += 4) {
        const int gb = t * 16 + mrow;
        for (int cot = 0; cot < 2; ++cot) {
            const int co = cot * 16 + mrow;
            v8f acc = {};
            int k = kof, ci0 = 0, kt0 = kof, ci1 = 0, kt1 = kof + 1;
            for (int ks = 0; ks < 40; ++ks) {
                v2f A, Bv;
                A.x = w2s[co][k];
                A.y = w2s[co][k + 1];
                int p0 = min(max(gb + kt0 - 2, 0), YG - 1);
                int p1 = min(max(gb + kt1 - 2, 0), YG - 1);
                Bv.x = y1s[ci0][p0];
                Bv.y = y1s[ci1][p1];
                acc = wmma_f32(A, Bv, acc);
                k += 4; kadv(ci0, kt0); kadv(ci1, kt1);
            }
#pragma unroll
            for (int r = 0; r < 8; ++r) {
                int c2 = cot * 16 + r + (hi ? 8 : 0);
                y2s[c2][t * 16 + mrow] = fmaxf(acc[r] + b2s[c2], 0.0f);
            }
        }
    }
    __syncthreads();

    // ---- layer 3: 32->2 (A rows >=2 zeroed by multiply) + select + store ----
    const float rowsel = (mrow < 2) ? 1.0f : 0.0f;   // zero A rows 2..15
    const int   wrow   = mrow & 1;                    // safe in-range row
    for (int ot = wave; ot < NTO; ot += 4) {
        const int gb = ot * 16 + mrow;
        v8f acc = {};
        int k = kof, ci0 = 0, kt0 = kof, ci1 = 0, kt1 = kof + 1;
        for (int ks = 0; ks < 40; ++ks) {
            v2f A, Bv;
            A.x  = rowsel * w3s[wrow][k];
            A.y  = rowsel * w3s[wrow][k + 1];
            Bv.x = y2s[ci0][gb + kt0 + 6];
            Bv.y = y2s[ci1][gb + kt1 + 6];
            acc = wmma_f32(A, Bv, acc);
            k += 4; kadv(ci0, kt0); kadv(ci1, kt1);
        }
        if (lane < 16) {  // C/D layout: M=0 -> acc[0] lanes 0-15, M=1 -> acc[1]
            int pos = n0 + ot * 16 + lane;
            float vi0 = acc[0] + b3s[0];
            float vi1 = acc[1] + b3s[1];
            float mk  = mask[row * N_DIM + pos];
            int xi    = ot * 16 + lane + 16;   // pos - (n0-16)
            float fr  = (mk > 0.5f) ? xs[0][xi] : vi0;
            float fi  = (mk > 0.5f) ? xs[1][xi] : vi1;
            int gi = row * N_DIM + pos;
            out[2 * gi + 0] = fr;
            out[2 * gi + 1] = fi;
        }
    }
}

// ============================================================================
extern "C" void kernel_launch(void* const* d_in, const int* in_sizes, int n_in,
                              void* d_out, int out_size, void* d_ws, size_t ws_size,
                              hipStream_t stream) {
    const float* v    = (const float*)d_in[0];
    const float* mask = (const float*)d_in[1];
    const float* w1   = (const float*)d_in[2];
    const float* b1   = (const float*)d_in[3];
    const float* w2   = (const float*)d_in[4];
    const float* b2   = (const float*)d_in[5];
    const float* w3   = (const float*)d_in[6];
    const float* b3   = (const float*)d_in[7];

    float* out        = (float*)d_out;                       // feats_final (B,N,2)
    float* out_sparse = out + (size_t)B_DIM * N_DIM * 2;     // feats_sparse (B,N,2)
    float* xws        = (float*)d_ws;                        // (B,2,N) scratch

    gii_scan_kernel<<<B_DIM, SCAN_T, 0, stream>>>(v, mask, out_sparse, xws);

    dim3 grid(N_DIM / TPOS, B_DIM);
    conv_interp_kernel<<<grid, 128, 0, stream>>>(mask, w1, b1, w2, b2, w3, b3,
                                                 xws, out);
}